// JointsLoss_sinkhorn_61211873902960
// MI455X (gfx1250) — compile-verified
//
#include <hip/hip_runtime.h>

// loss = 0.5 * mean( ((output - target) * target_weight)^2 )
// B=256, J=17, HW=64*48=3072  ->  N = 13,369,344 f32 elements per tensor.
// HBM-bound: ~107 MB @ 23.3 TB/s => ~4.6 us floor. Strategy:
//  - b128 non-temporal streaming loads (read-once data)
//  - per-thread f32 accumulation
//  - wave32 reduction via V_WMMA_F32_16X16X4_F32 (A = lane accs in K-column,
//    B = ones => D[m][n] = s_m + s_{m+16}), one shfl_xor(16) to finish
//  - deterministic two-kernel reduction (no float atomics)

typedef float v2f __attribute__((ext_vector_type(2)));
typedef float v4f __attribute__((ext_vector_type(4)));
typedef float v8f __attribute__((ext_vector_type(8)));

#define THREADS 256
#define NBLK    1280
#define HW4     768   // (64*48)/4 floats4 per (b,j) slice

__global__ __launch_bounds__(THREADS)
void ssd_partial_kernel(const float* __restrict__ pred,
                        const float* __restrict__ tgt,
                        const float* __restrict__ tw,
                        float* __restrict__ partial,
                        int n4)
{
    const int tid    = blockIdx.x * blockDim.x + threadIdx.x;
    const int stride = gridDim.x * blockDim.x;

    const v4f* __restrict__ p4 = (const v4f*)pred;
    const v4f* __restrict__ t4 = (const v4f*)tgt;

    float acc = 0.0f;
    for (int i = tid; i < n4; i += stride) {
        v4f o = __builtin_nontemporal_load(&p4[i]);   // global_load_b128 (NT)
        v4f t = __builtin_nontemporal_load(&t4[i]);
        float w = tw[i / HW4];                        // cached, reused 768x
        v4f d = o - t;
        float ss = d.x * d.x + d.y * d.y + d.z * d.z + d.w * d.w;
        acc = fmaf(w * w, ss, acc);
    }

    // ---- wave32 reduction with one full-precision f32 WMMA ----
    // A (16x4 f32): lanes 0-15 supply A[m][0]=acc_m (VGPR0) / A[m][1]=0,
    //               lanes 16-31 supply A[m][2]=acc_{m+16} / A[m][3]=0.
    // B (4x16)  : all ones.  D[m][n] = acc_m + acc_{m+16} for every column n.
    v2f a; a.x = acc;  a.y = 0.0f;
    v2f b; b.x = 1.0f; b.y = 1.0f;
    v8f c = {};
    v8f dm = __builtin_amdgcn_wmma_f32_16x16x4_f32(
        /*neg_a=*/false, a, /*neg_b=*/false, b,
        /*c_mod=*/(short)0, c, /*reuse_a=*/false, /*reuse_b=*/false);

    // lanes 0-15 hold M=0..7 pair-sums, lanes 16-31 hold M=8..15 pair-sums
    float half_sum = dm[0] + dm[1] + dm[2] + dm[3] + dm[4] + dm[5] + dm[6] + dm[7];
    float wave_sum = half_sum + __shfl_xor(half_sum, 16, 32);

    __shared__ float lds[THREADS / 32];
    const int lane = threadIdx.x & 31;
    const int wave = threadIdx.x >> 5;
    if (lane == 0) lds[wave] = wave_sum;
    __syncthreads();

    if (threadIdx.x == 0) {
        float s = 0.0f;
        #pragma unroll
        for (int w = 0; w < THREADS / 32; ++w) s += lds[w];  // fixed order
        partial[blockIdx.x] = s;
    }
}

__global__ __launch_bounds__(256)
void final_reduce_kernel(const float* __restrict__ partial,
                         float* __restrict__ out,
                         int nblk, float scale)
{
    __shared__ float lds[256];
    float s = 0.0f;
    for (int i = threadIdx.x; i < nblk; i += 256) s += partial[i]; // fixed order
    lds[threadIdx.x] = s;
    __syncthreads();
    #pragma unroll
    for (int off = 128; off > 0; off >>= 1) {
        if (threadIdx.x < (unsigned)off) lds[threadIdx.x] += lds[threadIdx.x + off];
        __syncthreads();
    }
    if (threadIdx.x == 0) out[0] = lds[0] * scale;
}

extern "C" void kernel_launch(void* const* d_in, const int* in_sizes, int n_in,
                              void* d_out, int out_size, void* d_ws, size_t ws_size,
                              hipStream_t stream)
{
    const float* pred = (const float*)d_in[0];   // output [B,J,H,W] f32
    const float* tgt  = (const float*)d_in[1];   // target [B,J,H,W] f32
    const float* tw   = (const float*)d_in[2];   // target_weight [B,J,1] f32

    const int N  = in_sizes[0];                  // 13,369,344 (divisible by 4)
    const int n4 = N / 4;

    float* partial = (float*)d_ws;               // NBLK floats of scratch

    ssd_partial_kernel<<<NBLK, THREADS, 0, stream>>>(pred, tgt, tw, partial, n4);
    final_reduce_kernel<<<1, 256, 0, stream>>>(partial, (float*)d_out,
                                               NBLK, 0.5f / (float)N);
}